// ManifoldBranchMixer_12240656794029
// MI455X (gfx1250) — compile-verified
//
#include <hip/hip_runtime.h>
#include <hip/hip_bf16.h>
#include <math.h>

// Problem constants (from the reference)
#define KB_ 4        // K branches
#define BB_ 8        // batch
#define TT_ 2048     // T
#define CC_ 1024     // C
#define HH_ 256      // HIDDEN
#define EPSV 1e-6f

typedef float v2f __attribute__((ext_vector_type(2)));
typedef float v8f __attribute__((ext_vector_type(8)));

// ---------------------------------------------------------------------------
// Kernel 1: zero the pooled accumulator (B*K*C = 32768 floats)
// ---------------------------------------------------------------------------
__global__ void mbm_zero_kernel(float* __restrict__ pooled) {
    int i = blockIdx.x * blockDim.x + threadIdx.x;
    if (i < BB_ * KB_ * CC_) pooled[i] = 0.0f;
}

// ---------------------------------------------------------------------------
// Kernel 2: pooling. pooled[b*K+k][c] += sum over a 128-row T-slab of
// branches[k][b][t][c]. Coalesced: 256 consecutive c per block row.
// grid = (K*B) * (C/256) * (T/128) = 32*4*16 = 2048 blocks, 256 threads.
// ---------------------------------------------------------------------------
__global__ void mbm_pool_kernel(const float* __restrict__ br,
                                float* __restrict__ pooled) {
    int bid  = blockIdx.x;
    int tblk = bid & 15;          // 16 T-slabs of 128 rows
    int cblk = (bid >> 4) & 3;    // 4 C-slabs of 256 cols
    int kb   = bid >> 6;          // 0..31 : k*B + b
    int c    = cblk * 256 + threadIdx.x;

    const float* base = br + ((size_t)kb * TT_ + (size_t)tblk * 128) * CC_ + c;
    float s = 0.0f;
    #pragma unroll 8
    for (int t = 0; t < 128; ++t) s += base[(size_t)t * CC_];

    int k = kb >> 3;              // kb = k*B + b, B=8
    int b = kb & 7;
    atomicAdd(&pooled[(b * KB_ + k) * CC_ + c], s);
}

// ---------------------------------------------------------------------------
// Kernel 3: h = tanh( (pooled/T) @ w_proj^T ) via V_WMMA_F32_16X16X4_F32.
// M = B*K = 32 rows, N = HIDDEN = 256 cols, Kdim = C = 1024.
// One wave (32 threads) per 16x16 output tile; grid = 2*16 = 32 blocks.
//
// ISA 16x4 f32 A layout: lanes 0-15 -> M=lane (K = 0,1 in VGPR 0,1),
//                        lanes 16-31 -> same M rows (K = 2,3 in VGPR 0,1).
// B (4x16): lane gives N, VGPR v gives K = 2*(lane>=16)+v (mirror of A).
// C/D (16x16 f32): VGPR i -> M = i + 8*(lane>=16), N = lane&15 (+tile).
// ---------------------------------------------------------------------------
__global__ void mbm_gemm_tanh_kernel(const float* __restrict__ pooled,  // (32,1024) sums
                                     const float* __restrict__ w_proj,  // (256,1024)
                                     float* __restrict__ h) {           // (32,256)
    const int mtile = blockIdx.x & 1;    // 2 M-tiles
    const int ntile = blockIdx.x >> 1;   // 16 N-tiles
    const int lane  = threadIdx.x;       // wave32, EXEC all ones
    const int r     = lane & 15;
    const int kh    = lane >> 4;         // 0 or 1: which K-half this lane holds

    const int arow = mtile * 16 + r;     // A row (b*K+k)
    const int ncol = ntile * 16 + r;     // D column / w_proj row

    const float* ap = pooled + (size_t)arow * CC_;
    const float* bp = w_proj + (size_t)ncol * CC_;

    v8f acc = {};
    for (int kb = 0; kb < CC_; kb += 4) {
        v2f a, b;
        a[0] = ap[kb + 2 * kh + 0];
        a[1] = ap[kb + 2 * kh + 1];
        b[0] = bp[kb + 2 * kh + 0];
        b[1] = bp[kb + 2 * kh + 1];
        acc = __builtin_amdgcn_wmma_f32_16x16x4_f32(
            /*neg_a=*/false, a, /*neg_b=*/false, b,
            /*c_mod=*/(short)0, acc, /*reuse_a=*/false, /*reuse_b=*/false);
    }

    const float invT = 1.0f / (float)TT_;   // pooled holds sums; mean = sum/T
    #pragma unroll
    for (int i = 0; i < 8; ++i) {
        int m = mtile * 16 + i + kh * 8;
        h[(size_t)m * HH_ + ncol] = tanhf(acc[i] * invT);
    }
}

// ---------------------------------------------------------------------------
// Kernel 4: logits + Sinkhorn -> weights (B*K = 32 floats). One block, 256 thr.
// ---------------------------------------------------------------------------
__global__ void mbm_weights_kernel(const float* __restrict__ h,      // (32,256)
                                   const float* __restrict__ w_out,  // (256)
                                   float* __restrict__ weights) {    // (32)
    __shared__ float logits_s[BB_ * KB_];
    const int tid  = threadIdx.x;
    const int wave = tid >> 5;
    const int lane = tid & 31;

    // 8 waves, each wave computes 4 of the 32 logits by wave-wide reduction
    for (int rr = 0; rr < 4; ++rr) {
        int row = wave * 4 + rr;
        float s = 0.0f;
        for (int j = lane; j < HH_; j += 32) s += h[row * HH_ + j] * w_out[j];
        #pragma unroll
        for (int off = 16; off > 0; off >>= 1) s += __shfl_xor(s, off, 32);
        if (lane == 0) logits_s[row] = s;
    }
    __syncthreads();

    // Threads 0..7: per-batch 4x4 Sinkhorn, fully in registers.
    if (tid < BB_) {
        float lg[KB_];
        #pragma unroll
        for (int i = 0; i < KB_; ++i) lg[i] = logits_s[tid * KB_ + i];

        float mx = -INFINITY;
        #pragma unroll
        for (int i = 0; i < KB_; ++i)
            for (int j = 0; j < KB_; ++j) mx = fmaxf(mx, lg[i] + lg[j]);

        float t[KB_][KB_];
        #pragma unroll
        for (int i = 0; i < KB_; ++i)
            for (int j = 0; j < KB_; ++j)
                t[i][j] = fmaxf(__expf(lg[i] + lg[j] - mx), EPSV);

        for (int it = 0; it < 5; ++it) {
            #pragma unroll
            for (int i = 0; i < KB_; ++i) {
                float rs = t[i][0] + t[i][1] + t[i][2] + t[i][3];
                float inv = 1.0f / (rs + EPSV);
                for (int j = 0; j < KB_; ++j) t[i][j] *= inv;
            }
            #pragma unroll
            for (int j = 0; j < KB_; ++j) {
                float cs = t[0][j] + t[1][j] + t[2][j] + t[3][j];
                float inv = 1.0f / (cs + EPSV);
                for (int i = 0; i < KB_; ++i) t[i][j] *= inv;
            }
        }
        #pragma unroll
        for (int i = 0; i < KB_; ++i) {
            float rs = t[i][0] + t[i][1] + t[i][2] + t[i][3];
            float inv = 1.0f / (rs + EPSV);
            for (int j = 0; j < KB_; ++j) t[i][j] *= inv;
        }

        float w[KB_];
        #pragma unroll
        for (int k = 0; k < KB_; ++k)
            w[k] = 0.25f * (t[0][k] + t[1][k] + t[2][k] + t[3][k]);
        float wsum = w[0] + w[1] + w[2] + w[3];
        float inv = 1.0f / (wsum + EPSV);
        bool valid = true;
        #pragma unroll
        for (int k = 0; k < KB_; ++k) {
            w[k] *= inv;
            valid = valid && isfinite(w[k]);
        }
        #pragma unroll
        for (int k = 0; k < KB_; ++k)
            weights[tid * KB_ + k] = valid ? w[k] : 0.25f;
    }
}

// ---------------------------------------------------------------------------
// Kernel 5: mix. out[b][t][c] = sum_k w[b][k] * branches[k][b][t][c].
// float4-vectorized streaming; T*C/4 = 524288 is a multiple of blockDim, so
// every block sees a single b. grid = B*T*C/4/256 = 16384 blocks.
// ---------------------------------------------------------------------------
__global__ void mbm_mix_kernel(const float* __restrict__ br,
                               const float* __restrict__ weights,
                               float* __restrict__ out) {
    const size_t TC4 = (size_t)TT_ * CC_ / 4;
    size_t idx = (size_t)blockIdx.x * blockDim.x + threadIdx.x;
    int b = (int)(idx / TC4);
    size_t rem = idx % TC4;

    float w0 = weights[b * KB_ + 0];
    float w1 = weights[b * KB_ + 1];
    float w2 = weights[b * KB_ + 2];
    float w3 = weights[b * KB_ + 3];

    const size_t planeTC = (size_t)TT_ * CC_;
    const float4* p0 = (const float4*)(br + ((size_t)0 * BB_ + b) * planeTC) + rem;
    const float4* p1 = (const float4*)(br + ((size_t)1 * BB_ + b) * planeTC) + rem;
    const float4* p2 = (const float4*)(br + ((size_t)2 * BB_ + b) * planeTC) + rem;
    const float4* p3 = (const float4*)(br + ((size_t)3 * BB_ + b) * planeTC) + rem;

    float4 x0 = *p0, x1 = *p1, x2 = *p2, x3 = *p3;
    float4 r;
    r.x = w0 * x0.x + w1 * x1.x + w2 * x2.x + w3 * x3.x;
    r.y = w0 * x0.y + w1 * x1.y + w2 * x2.y + w3 * x3.y;
    r.z = w0 * x0.z + w1 * x1.z + w2 * x2.z + w3 * x3.z;
    r.w = w0 * x0.w + w1 * x1.w + w2 * x2.w + w3 * x3.w;

    ((float4*)(out + (size_t)b * planeTC))[rem] = r;
}

// ---------------------------------------------------------------------------
extern "C" void kernel_launch(void* const* d_in, const int* in_sizes, int n_in,
                              void* d_out, int out_size, void* d_ws, size_t ws_size,
                              hipStream_t stream) {
    const float* branches = (const float*)d_in[0];  // (K,B,T,C) f32
    const float* w_proj   = (const float*)d_in[1];  // (HIDDEN,C) f32
    const float* w_out    = (const float*)d_in[2];  // (1,HIDDEN) f32
    float* out = (float*)d_out;                     // (B,T,C) f32

    // Workspace layout (floats)
    float* ws      = (float*)d_ws;
    float* pooled  = ws;                         // 32 * 1024
    float* h       = ws + BB_ * KB_ * CC_;       // 32 * 256
    float* weights = h + BB_ * KB_ * HH_;        // 32

    // 1) zero pooled accumulator
    mbm_zero_kernel<<<(BB_ * KB_ * CC_ + 255) / 256, 256, 0, stream>>>(pooled);

    // 2) pooling: (K*B) * (C/256) * (T/128) blocks
    mbm_pool_kernel<<<KB_ * BB_ * 4 * 16, 256, 0, stream>>>(branches, pooled);

    // 3) WMMA f32 projection + tanh: 2 M-tiles * 16 N-tiles, one wave each
    mbm_gemm_tanh_kernel<<<32, 32, 0, stream>>>(pooled, w_proj, h);

    // 4) logits + sinkhorn -> per-batch weights
    mbm_weights_kernel<<<1, 256, 0, stream>>>(h, w_out, weights);

    // 5) weighted blend (bandwidth-dominant)
    const size_t n4 = (size_t)BB_ * TT_ * CC_ / 4;
    mbm_mix_kernel<<<(unsigned)(n4 / 256), 256, 0, stream>>>(branches, weights, out);
}